// MIGNNv3_80788334837818
// MI455X (gfx1250) — compile-verified
//
#include <hip/hip_runtime.h>
#include <hip/hip_bf16.h>
#include <stdint.h>

#define NNODES 100000
#define NEDGES 600000
#define DIM 128
#define SCAN_CHUNK 1024
#define NB ((NNODES + SCAN_CHUNK - 1) / SCAN_CHUNK)   // 98 scan blocks
#define ROWPAD 132                                     // LDS row stride (floats) -> conflict-free

typedef float v2f __attribute__((ext_vector_type(2)));
typedef float v4f __attribute__((ext_vector_type(4)));
typedef float v8f __attribute__((ext_vector_type(8)));

// ---------- CSR construction (counting sort, deterministic) ----------

__global__ void k_zero(int* __restrict__ p, int n) {
  int i = blockIdx.x * blockDim.x + threadIdx.x;
  if (i < n) p[i] = 0;
}

__global__ void k_count(const int* __restrict__ ei, int* __restrict__ deg) {
  int e = blockIdx.x * blockDim.x + threadIdx.x;
  if (e < NEDGES) atomicAdd(&deg[ei[NEDGES + e]], 1);   // row 1 of edge_index = dst
}

__global__ void k_scan_partial(const int* __restrict__ deg, int* __restrict__ bsum) {
  __shared__ int s[256];
  int t = threadIdx.x;
  int base = blockIdx.x * SCAN_CHUNK + t * 4;
  int sum = 0;
  #pragma unroll
  for (int j = 0; j < 4; ++j) { int idx = base + j; sum += (idx < NNODES) ? deg[idx] : 0; }
  s[t] = sum;
  __syncthreads();
  for (int d = 128; d > 0; d >>= 1) {
    if (t < d) s[t] += s[t + d];
    __syncthreads();
  }
  if (t == 0) bsum[blockIdx.x] = s[0];
}

__global__ void k_scan_blocksums(const int* __restrict__ bsum, int* __restrict__ boff,
                                 int* __restrict__ offsets) {
  __shared__ int s[128];
  int t = threadIdx.x;
  int own = (t < NB) ? bsum[t] : 0;
  s[t] = own;
  __syncthreads();
  for (int d = 1; d < 128; d <<= 1) {
    int add = (t >= d) ? s[t - d] : 0;
    __syncthreads();
    s[t] += add;
    __syncthreads();
  }
  if (t < NB) boff[t] = s[t] - own;            // exclusive prefix of block sums
  if (t == 0) offsets[NNODES] = NEDGES;        // total degree == E by construction
}

__global__ void k_scan_final(const int* __restrict__ deg, const int* __restrict__ boff,
                             int* __restrict__ offsets) {
  __shared__ int s[256];
  int t = threadIdx.x;
  int base = blockIdx.x * SCAN_CHUNK + t * 4;
  int v[4]; int tsum = 0;
  #pragma unroll
  for (int j = 0; j < 4; ++j) { int idx = base + j; v[j] = (idx < NNODES) ? deg[idx] : 0; tsum += v[j]; }
  s[t] = tsum;
  __syncthreads();
  for (int d = 1; d < 256; d <<= 1) {
    int add = (t >= d) ? s[t - d] : 0;
    __syncthreads();
    s[t] += add;
    __syncthreads();
  }
  int running = boff[blockIdx.x] + (s[t] - tsum);   // exclusive prefix for this thread
  #pragma unroll
  for (int j = 0; j < 4; ++j) {
    int idx = base + j;
    if (idx < NNODES) offsets[idx] = running;
    running += v[j];
  }
}

__global__ void k_place(const int* __restrict__ ei, const int* __restrict__ offsets,
                        int* __restrict__ cursor, int* __restrict__ csr) {
  int e = blockIdx.x * blockDim.x + threadIdx.x;
  if (e < NEDGES) {
    int d = ei[NEDGES + e];
    int p = atomicAdd(&cursor[d], 1);
    csr[offsets[d] + p] = e;                   // store edge id (sorted next for determinism)
  }
}

__global__ void k_sort_srcify(const int* __restrict__ ei, const int* __restrict__ offsets,
                              int* __restrict__ csr) {
  int n = blockIdx.x * blockDim.x + threadIdx.x;
  if (n >= NNODES) return;
  int s = offsets[n], t = offsets[n + 1];
  for (int i = s + 1; i < t; ++i) {            // insertion sort by edge id (avg degree 6)
    int key = csr[i]; int j = i - 1;
    while (j >= s && csr[j] > key) { csr[j + 1] = csr[j]; --j; }
    csr[j + 1] = key;
  }
  for (int i = s; i < t; ++i) csr[i] = ei[csr[i]];   // edge id -> src node id
}

// ---------- fused: stage x-tile, segment-product tile, dual fp32 WMMA GEMM ----------

__global__ __launch_bounds__(128)
void k_fused(const float* __restrict__ x,
             const float* __restrict__ W1, const float* __restrict__ b1,
             const float* __restrict__ W2, const float* __restrict__ b2,
             const int* __restrict__ offsets, const int* __restrict__ csr,
             float* __restrict__ out) {
  __shared__ float xs[16 * ROWPAD];   // x tile,   16 nodes x 128 (padded)
  __shared__ float as[16 * ROWPAD];   // aggr tile, 16 nodes x 128 (padded)
  const int tid = threadIdx.x;
  const int n0 = blockIdx.x * 16;

  // phase 1: stage node-feature tile (8 threads per node, 16 floats each)
  {
    int r = tid >> 3;
    int c0 = (tid & 7) * 16;
    const v4f* sp = (const v4f*)(x + (size_t)(n0 + r) * DIM + c0);
    v4f* dp = (v4f*)(xs + r * ROWPAD + c0);
    dp[0] = sp[0]; dp[1] = sp[1]; dp[2] = sp[2]; dp[3] = sp[3];
  }

  // phase 2: segment product over incoming edges (identity 1); gathers are L2-resident
  {
    int r = tid >> 3;
    int c0 = (tid & 7) * 16;
    int n = n0 + r;
    v4f p0 = {1.f,1.f,1.f,1.f}, p1 = p0, p2 = p0, p3 = p0;
    int beg = offsets[n], end = offsets[n + 1];
    for (int i = beg; i < end; ++i) {
      int srcn = csr[i];
      const v4f* sv = (const v4f*)(x + (size_t)srcn * DIM + c0);
      p0 *= sv[0]; p1 *= sv[1]; p2 *= sv[2]; p3 *= sv[3];
    }
    v4f* dp = (v4f*)(as + r * ROWPAD + c0);
    dp[0] = p0; dp[1] = p1; dp[2] = p2; dp[3] = p3;
  }
  __syncthreads();

  // phase 3: h1 = x@W1^T, h2 = aggr@W2^T via V_WMMA_F32_16X16X4_F32; out = (h1+b1)*(h2+b2)
  const int wave = tid >> 5;
  const int lane = tid & 31;
  const int r16  = lane & 15;
  const int hi   = lane >> 4;          // lanes 16-31 carry K+2/K+3 (A) and M+8 (C/D)

  for (int jj = 0; jj < 2; ++jj) {     // each wave owns two 16-wide output column tiles
    int j0 = (wave * 2 + jj) * 16;
    int col = j0 + r16;
    v8f acc1 = {0.f,0.f,0.f,0.f,0.f,0.f,0.f,0.f};
    v8f acc2 = {0.f,0.f,0.f,0.f,0.f,0.f,0.f,0.f};
    const float* w1p = W1 + (size_t)col * DIM;   // B(k, col) = W[col*128 + k]
    const float* w2p = W2 + (size_t)col * DIM;
    const float* xr  = xs + r16 * ROWPAD;
    const float* ar  = as + r16 * ROWPAD;
    for (int k0 = 0; k0 < DIM; k0 += 4) {
      int ka = k0 + 2 * hi;
      v2f a1  = *(const v2f*)(xr + ka);
      v2f a2  = *(const v2f*)(ar + ka);
      v2f bw1 = *(const v2f*)(w1p + ka);
      v2f bw2 = *(const v2f*)(w2p + ka);
      acc1 = __builtin_amdgcn_wmma_f32_16x16x4_f32(false, a1, false, bw1,
                                                   (short)0, acc1, false, false);
      acc2 = __builtin_amdgcn_wmma_f32_16x16x4_f32(false, a2, false, bw2,
                                                   (short)0, acc2, false, false);
    }
    float bias1 = b1[col], bias2 = b2[col];
    #pragma unroll
    for (int i = 0; i < 8; ++i) {      // C/D layout: VGPR i -> M = i + 8*hi, N = r16
      int m = i + 8 * hi;
      out[(size_t)(n0 + m) * DIM + col] = (acc1[i] + bias1) * (acc2[i] + bias2);
    }
  }
}

// ---------- launcher ----------

extern "C" void kernel_launch(void* const* d_in, const int* in_sizes, int n_in,
                              void* d_out, int out_size, void* d_ws, size_t ws_size,
                              hipStream_t stream) {
  (void)in_sizes; (void)n_in; (void)out_size; (void)ws_size;
  const float* x  = (const float*)d_in[0];
  const int*   ei = (const int*)d_in[1];
  const float* W1 = (const float*)d_in[2];
  const float* b1 = (const float*)d_in[3];
  const float* W2 = (const float*)d_in[4];
  const float* b2 = (const float*)d_in[5];
  float* out = (float*)d_out;

  int* ws      = (int*)d_ws;
  int* offsets = ws;                        // N+1
  int* deg     = offsets + (NNODES + 1);    // N
  int* cursor  = deg + NNODES;              // N (contiguous with deg -> one zero pass)
  int* bsum    = cursor + NNODES;           // NB
  int* boff    = bsum + NB;                 // NB
  int* csr     = boff + NB;                 // E

  k_zero<<<(2 * NNODES + 255) / 256, 256, 0, stream>>>(deg, 2 * NNODES);
  k_count<<<(NEDGES + 255) / 256, 256, 0, stream>>>(ei, deg);
  k_scan_partial<<<NB, 256, 0, stream>>>(deg, bsum);
  k_scan_blocksums<<<1, 128, 0, stream>>>(bsum, boff, offsets);
  k_scan_final<<<NB, 256, 0, stream>>>(deg, boff, offsets);
  k_place<<<(NEDGES + 255) / 256, 256, 0, stream>>>(ei, offsets, cursor, csr);
  k_sort_srcify<<<(NNODES + 255) / 256, 256, 0, stream>>>(ei, offsets, csr);
  k_fused<<<NNODES / 16, 128, 0, stream>>>(x, W1, b1, W2, b2, offsets, csr, out);
}